// MeasureDistance_24584392802815
// MI455X (gfx1250) — compile-verified
//
#include <hip/hip_runtime.h>
#include <math.h>

typedef __attribute__((ext_vector_type(2))) float v2f;
typedef __attribute__((ext_vector_type(8))) float v8f;

#define B_     8
#define NPT    2048         // L == Kn
#define D_     32
#define NB     (B_ * NPT)   // 16384
#define NCHUNK 16           // row-chunks for the column (transposed) reduction

// ---------------- norms: ||x_i||^2 and ||y_j||^2 ----------------
__global__ void norms_kernel(const float* __restrict__ X, const float* __restrict__ Y,
                             float* __restrict__ nx, float* __restrict__ ny) {
  int idx = blockIdx.x * blockDim.x + threadIdx.x;   // [0, 2*NB)
  const float* src;
  float* dst;
  int i;
  if (idx < NB) { src = X; dst = nx; i = idx; }
  else          { src = Y; dst = ny; i = idx - NB; }
  const float* p = src + (size_t)i * D_;
  float s = 0.f;
#pragma unroll
  for (int d = 0; d < D_; ++d) s += p[d] * p[d];
  dst[i] = s;
}

// ---------------- kernel matrix build via V_WMMA_F32_16X16X4_F32 ----------------
// K[m][n] = -max(||x_m||^2 + ||y_n||^2 - 2 x_m.y_n, 0) / 2   (sigma = 1)
__global__ void build_kernel_wmma(const float* __restrict__ X, const float* __restrict__ Y,
                                  const float* __restrict__ nx, const float* __restrict__ ny,
                                  float* __restrict__ Kout) {
  int b = blockIdx.z;
  const float* Xb  = X + (size_t)b * NPT * D_;
  const float* Yb  = Y + (size_t)b * NPT * D_;
  const float* nxb = nx + (size_t)b * NPT;
  const float* nyb = ny + (size_t)b * NPT;
  float* Ko = Kout + (size_t)b * NPT * NPT;

  int lane = threadIdx.x & 31;
  int half = lane >> 4;
  int lm   = lane & 15;
  int m0 = blockIdx.y * 16;
  int n0 = blockIdx.x * 16;

  v8f c = {};
#pragma unroll
  for (int s = 0; s < 8; ++s) {
    int d = 4 * s + 2 * half;
    v2f aa = *(const v2f*)(Xb + (size_t)(m0 + lm) * D_ + d);
    v2f bb = *(const v2f*)(Yb + (size_t)(n0 + lm) * D_ + d);
    c = __builtin_amdgcn_wmma_f32_16x16x4_f32(false, aa, false, bb,
                                              (short)0, c, false, false);
  }

  float nyv = nyb[n0 + lm];
#pragma unroll
  for (int j = 0; j < 8; ++j) {
    int m = m0 + j + 8 * half;
    float val = nxb[m] + nyv - 2.0f * c[j];
    val = fmaxf(val, 0.0f) * 0.5f;
    Ko[(size_t)m * NPT + (n0 + lm)] = -val;
  }
}

// ---------------- fused row logsumexp over 3 matrices ----------------
// out[i] = -LSE_j(K[i][j] + q[j]) ; q = pot + w precombined.
// One wave32 per row; b128 streaming loads; branchless group-of-4 online LSE.
struct RowArgs {
  const float* K[3];
  const float* q[3];      // reduction pot + weight, (B, NPT)
  const float* selfp[3];  // old self potential, (B, NPT)
  const float* w[3];      // companion weight (to emit q_out)
  float* potOut[3];
  float* qOut[3];
  int avg;
};

__global__ void lse_row_fused(RowArgs ar) {
  int wid  = (blockIdx.x * blockDim.x + threadIdx.x) >> 5;  // [0, 3*NB)
  int lane = threadIdx.x & 31;
  int mtx  = wid / NB;
  int rid  = wid - mtx * NB;               // [0, NB): b*NPT + i
  const float* Kr = ar.K[mtx] + (size_t)rid * NPT;
  const float* qv = ar.q[mtx] + (size_t)(rid / NPT) * NPT;

  float m = -INFINITY, s = 0.0f;
  for (int t = lane * 4; t < NPT; t += 128) {
    float4 k4 = *(const float4*)(Kr + t);
    float4 q4 = *(const float4*)(qv + t);
    float v0 = k4.x + q4.x, v1 = k4.y + q4.y;
    float v2 = k4.z + q4.z, v3 = k4.w + q4.w;
    float vm = fmaxf(fmaxf(v0, v1), fmaxf(v2, v3));
    float nm = fmaxf(m, vm);
    s = s * __expf(m - nm) + __expf(v0 - nm) + __expf(v1 - nm)
                           + __expf(v2 - nm) + __expf(v3 - nm);
    m = nm;
  }
#pragma unroll
  for (int off = 16; off > 0; off >>= 1) {
    float om = __shfl_xor(m, off, 32);
    float os = __shfl_xor(s, off, 32);
    float nm = fmaxf(m, om);
    s = s * __expf(m - nm) + os * __expf(om - nm);
    m = nm;
  }
  if (lane == 0) {
    float r = -(m + __logf(s));
    if (ar.avg) {
      float pn = 0.5f * (r + ar.selfp[mtx][rid]);
      ar.potOut[mtx][rid] = pn;
      ar.qOut[mtx][rid]   = pn + ar.w[mtx][rid];
    } else {
      ar.potOut[mtx][rid] = r;
    }
  }
}

// ---------------- column logsumexp over Kxy (rows of Kyx) ----------------
// Phase 1: each thread owns 4 consecutive columns (b128 loads, coalesced),
// one 128-row chunk; branchless group-of-4-rows online LSE per column.
__global__ void lse_col_partial(const float* __restrict__ Kmat,
                                const float* __restrict__ q,
                                float* __restrict__ pm, float* __restrict__ ps) {
  int c0    = (blockIdx.x * blockDim.x + threadIdx.x) * 4;  // [0, NPT)
  int chunk = blockIdx.y;
  int b     = blockIdx.z;
  const float* Kb = Kmat + (size_t)b * NPT * NPT;
  const float* qv = q + (size_t)b * NPT;
  int j0 = chunk * (NPT / NCHUNK);

  float m[4] = {-INFINITY, -INFINITY, -INFINITY, -INFINITY};
  float s[4] = {0.f, 0.f, 0.f, 0.f};
  for (int j = j0; j < j0 + NPT / NCHUNK; j += 4) {
    float4 r0 = *(const float4*)(Kb + (size_t)(j + 0) * NPT + c0);
    float4 r1 = *(const float4*)(Kb + (size_t)(j + 1) * NPT + c0);
    float4 r2 = *(const float4*)(Kb + (size_t)(j + 2) * NPT + c0);
    float4 r3 = *(const float4*)(Kb + (size_t)(j + 3) * NPT + c0);
    float q0 = qv[j], q1 = qv[j + 1], q2 = qv[j + 2], q3 = qv[j + 3];
    float v[4][4] = {
      {r0.x + q0, r1.x + q1, r2.x + q2, r3.x + q3},
      {r0.y + q0, r1.y + q1, r2.y + q2, r3.y + q3},
      {r0.z + q0, r1.z + q1, r2.z + q2, r3.z + q3},
      {r0.w + q0, r1.w + q1, r2.w + q2, r3.w + q3}};
#pragma unroll
    for (int c = 0; c < 4; ++c) {
      float vm = fmaxf(fmaxf(v[c][0], v[c][1]), fmaxf(v[c][2], v[c][3]));
      float nm = fmaxf(m[c], vm);
      s[c] = s[c] * __expf(m[c] - nm) + __expf(v[c][0] - nm) + __expf(v[c][1] - nm)
                                      + __expf(v[c][2] - nm) + __expf(v[c][3] - nm);
      m[c] = nm;
    }
  }
  size_t pidx = ((size_t)(b * NCHUNK + chunk)) * NPT + c0;
  *(float4*)(pm + pidx) = make_float4(m[0], m[1], m[2], m[3]);
  *(float4*)(ps + pidx) = make_float4(s[0], s[1], s[2], s[3]);
}

// Phase 2: merge (m,s) partials per column; emit pot (and q when averaging).
__global__ void lse_col_combine(const float* __restrict__ pm, const float* __restrict__ ps,
                                const float* __restrict__ selfp, const float* __restrict__ w,
                                float* __restrict__ potOut, float* __restrict__ qOut,
                                int avg) {
  int col = blockIdx.x * blockDim.x + threadIdx.x;
  int b   = blockIdx.y;
  float m = -INFINITY, s = 0.0f;
#pragma unroll
  for (int c = 0; c < NCHUNK; ++c) {
    size_t pidx = ((size_t)(b * NCHUNK + c)) * NPT + col;
    float cm = pm[pidx], cs = ps[pidx];
    float nm = fmaxf(m, cm);
    s = s * __expf(m - nm) + cs * __expf(cm - nm);
    m = nm;
  }
  float r = -(m + __logf(s));
  size_t o = (size_t)b * NPT + col;
  if (avg) {
    float pn = 0.5f * (r + selfp[o]);
    potOut[o] = pn;
    qOut[o]   = pn + w[o];
  } else {
    potOut[o] = r;
  }
}

// ---------------- init: pots = 0, q = companion weight ----------------
__global__ void init_pots(const float* __restrict__ a, const float* __restrict__ bw,
                          float* __restrict__ P, float* __restrict__ Q) {
  int i = blockIdx.x * blockDim.x + threadIdx.x;   // [0, NB)
  P[0 * NB + i] = 0.f;  // fxx
  P[1 * NB + i] = 0.f;  // fyy
  P[2 * NB + i] = 0.f;  // fxy
  P[3 * NB + i] = 0.f;  // fyx
  float av = a[i], bv = bw[i];
  Q[0 * NB + i] = av;   // qfxx = fxx + wa
  Q[1 * NB + i] = bv;   // qfyy = fyy + wb
  Q[2 * NB + i] = av;   // qfxy = fxy + wa
  Q[3 * NB + i] = bv;   // qfyx = fyx + wb
}

// ---------------- final weighted reduction ----------------
__global__ void final_reduce(const float* __restrict__ nxx, const float* __restrict__ nyy,
                             const float* __restrict__ nxy, const float* __restrict__ nyx,
                             const float* __restrict__ aw, const float* __restrict__ bw,
                             float* __restrict__ outp) {
  __shared__ float red[256];
  int b = blockIdx.x, t = threadIdx.x;
  size_t base = (size_t)b * NPT;
  float acc = 0.f;
  for (int i = t; i < NPT; i += 256) {
    acc += (nxy[base + i] - nxx[base + i]) * expf(aw[base + i]);
    acc += (nyx[base + i] - nyy[base + i]) * expf(bw[base + i]);
  }
  red[t] = acc;
  __syncthreads();
  for (int off = 128; off > 0; off >>= 1) {
    if (t < off) red[t] += red[t + off];
    __syncthreads();
  }
  if (t == 0) outp[b] = red[0];
}

// ---------------- host launch ----------------
extern "C" void kernel_launch(void* const* d_in, const int* in_sizes, int n_in,
                              void* d_out, int out_size, void* d_ws, size_t ws_size,
                              hipStream_t stream) {
  (void)in_sizes; (void)n_in; (void)out_size; (void)ws_size;
  const float* x  = (const float*)d_in[0];
  const float* a  = (const float*)d_in[1];
  const float* y  = (const float*)d_in[2];
  const float* bw = (const float*)d_in[3];
  float* out = (float*)d_out;

  float* ws = (float*)d_ws;
  const size_t MN = (size_t)NPT * NPT;
  float* Kxx = ws;
  float* Kyy = Kxx + (size_t)B_ * MN;
  float* Kxy = Kyy + (size_t)B_ * MN;
  float* nx  = Kxy + (size_t)B_ * MN;
  float* ny  = nx + NB;
  float* PP0 = ny + NB;                 // [0..3]=pots, [4..7]=q  (8*NB)
  float* PP1 = PP0 + 8 * (size_t)NB;
  float* FIN = PP1 + 8 * (size_t)NB;    // 4*NB
  float* PM  = FIN + 4 * (size_t)NB;    // B*NCHUNK*NPT
  float* PS  = PM + (size_t)B_ * NCHUNK * NPT;

  norms_kernel<<<dim3(2 * NB / 256), dim3(256), 0, stream>>>(x, y, nx, ny);

  dim3 bgrid(NPT / 16, NPT / 16, B_);
  build_kernel_wmma<<<bgrid, dim3(32), 0, stream>>>(x, x, nx, nx, Kxx);
  build_kernel_wmma<<<bgrid, dim3(32), 0, stream>>>(y, y, ny, ny, Kyy);
  build_kernel_wmma<<<bgrid, dim3(32), 0, stream>>>(x, y, nx, ny, Kxy);

  init_pots<<<dim3(NB / 256), dim3(256), 0, stream>>>(a, bw, PP0, PP0 + 4 * (size_t)NB);

  float* cur = PP0;
  float* nxt = PP1;
  dim3 rgrid(3 * NB / 8), rblk(256);
  dim3 cpgrid(NPT / 1024, NCHUNK, B_);
  dim3 ccgrid(NPT / 256, B_);

  for (int it = 0; it < 10; ++it) {
    float* cp = cur;               // pots fxx,fyy,fxy,fyx
    float* cq = cur + 4 * (size_t)NB;  // q    qfxx,qfyy,qfxy,qfyx
    float* np = nxt;
    float* nq = nxt + 4 * (size_t)NB;

    RowArgs ar;
    ar.K[0] = Kxx; ar.q[0] = cq + 0 * NB; ar.selfp[0] = cp + 0 * NB; ar.w[0] = a;
    ar.potOut[0] = np + 0 * NB; ar.qOut[0] = nq + 0 * NB;
    ar.K[1] = Kyy; ar.q[1] = cq + 1 * NB; ar.selfp[1] = cp + 1 * NB; ar.w[1] = bw;
    ar.potOut[1] = np + 1 * NB; ar.qOut[1] = nq + 1 * NB;
    ar.K[2] = Kxy; ar.q[2] = cq + 3 * NB; ar.selfp[2] = cp + 2 * NB; ar.w[2] = a;
    ar.potOut[2] = np + 2 * NB; ar.qOut[2] = nq + 2 * NB;
    ar.avg = 1;
    lse_row_fused<<<rgrid, rblk, 0, stream>>>(ar);

    lse_col_partial<<<cpgrid, dim3(256), 0, stream>>>(Kxy, cq + 2 * NB, PM, PS);
    lse_col_combine<<<ccgrid, dim3(256), 0, stream>>>(PM, PS, cp + 3 * NB, bw,
                                                      np + 3 * NB, nq + 3 * NB, 1);
    float* t = cur; cur = nxt; nxt = t;
  }

  // final extrapolation (no averaging)
  {
    float* cp = cur;
    float* cq = cur + 4 * (size_t)NB;
    (void)cp;
    RowArgs ar;
    ar.K[0] = Kxx; ar.q[0] = cq + 0 * NB; ar.selfp[0] = nullptr; ar.w[0] = nullptr;
    ar.potOut[0] = FIN + 0 * NB; ar.qOut[0] = nullptr;
    ar.K[1] = Kyy; ar.q[1] = cq + 1 * NB; ar.selfp[1] = nullptr; ar.w[1] = nullptr;
    ar.potOut[1] = FIN + 1 * NB; ar.qOut[1] = nullptr;
    ar.K[2] = Kxy; ar.q[2] = cq + 3 * NB; ar.selfp[2] = nullptr; ar.w[2] = nullptr;
    ar.potOut[2] = FIN + 2 * NB; ar.qOut[2] = nullptr;
    ar.avg = 0;
    lse_row_fused<<<rgrid, rblk, 0, stream>>>(ar);

    lse_col_partial<<<cpgrid, dim3(256), 0, stream>>>(Kxy, cq + 2 * NB, PM, PS);
    lse_col_combine<<<ccgrid, dim3(256), 0, stream>>>(PM, PS, nullptr, nullptr,
                                                      FIN + 3 * NB, nullptr, 0);
  }

  final_reduce<<<dim3(B_), dim3(256), 0, stream>>>(FIN + 0 * NB, FIN + 1 * NB,
                                                   FIN + 2 * NB, FIN + 3 * NB,
                                                   a, bw, out);
}